// EpsThresholdModel_62010737819654
// MI455X (gfx1250) — compile-verified
//
#include <hip/hip_runtime.h>
#include <cstdint>
#include <cstddef>

// EpsThreshold: per-row eps-threshold selection (cumsum-crossing + rank gather).
// One workgroup per row. Entire 50257-float row is LDS-resident (CDNA5: 320KB/WGP),
// staged via async global->LDS DMA (ASYNCcnt), then 12+10+10-bit radix-select.

#define EPS_D 0.1

// LDS layout (bytes)
#define OFF_HSUM   0u        // float  [4096]
#define OFF_HCNT   16384u    // u32    [4096]
#define OFF_HCNT0  32768u    // u32    [4096]  (saved round-0 counts, reused by stage 2)
#define OFF_GSUM   49152u    // double [128]
#define OFF_GCNT   50176u    // u32    [128]
#define OFF_SCRU   50688u    // u32    [16]
#define OFF_SCRD   50752u    // double [8]
#define OFF_VALS   50816u    // u32    [V]   (16B aligned)

// Find first bin where baseSum + inclusive f32-bin-sum crosses EPS (two-level scan).
// Results: scr_u[0]=bin, scr_u[1]=count in bin, scr_d[0]=sum strictly below bin.
__device__ __forceinline__ void scan_cross_sum(const float* hsum, const uint32_t* hcnt,
                                               double* gsum, uint32_t* gcnt,
                                               uint32_t* scr_u, double* scr_d,
                                               int NB, double baseSum, int tid) {
    const int NG = NB >> 5;
    if (tid < NG) {
        double s = 0.0; uint32_t c = 0u;
        const int b0 = tid << 5;
        for (int b = b0; b < b0 + 32; ++b) { s += (double)hsum[b]; c += hcnt[b]; }
        gsum[tid] = s; gcnt[tid] = c;
    }
    __syncthreads();
    if (tid == 0) {
        double s = baseSum; int g = NG - 1;
        for (int t = 0; t < NG; ++t) {
            double ns = s + gsum[t];
            if (gcnt[t] != 0u && ns > EPS_D) { g = t; break; }
            s = ns;
        }
        uint32_t B = (uint32_t)((g << 5) + 31);
        for (int b = (g << 5); b < (g << 5) + 32; ++b) {
            double ns = s + (double)hsum[b];
            if (hcnt[b] != 0u && ns > EPS_D) { B = (uint32_t)b; break; }
            s = ns;
        }
        scr_u[0] = B; scr_u[1] = hcnt[B]; scr_d[0] = s;
    }
    __syncthreads();
}

// Find first bin where baseCnt + inclusive count exceeds rank j (two-level scan).
// Results: scr_u[2]=bin, scr_u[3]=count strictly below bin.
__device__ __forceinline__ void scan_rank(const uint32_t* hc, uint32_t* gcnt,
                                          uint32_t* scr_u, int NB,
                                          uint32_t baseCnt, uint32_t jrank, int tid) {
    const int NG = NB >> 5;
    if (tid < NG) {
        uint32_t c = 0u; const int b0 = tid << 5;
        for (int b = b0; b < b0 + 32; ++b) c += hc[b];
        gcnt[tid] = c;
    }
    __syncthreads();
    if (tid == 0) {
        uint32_t cnt = baseCnt; int g = NG - 1;
        for (int t = 0; t < NG; ++t) {
            uint32_t nc = cnt + gcnt[t];
            if (nc > jrank) { g = t; break; }
            cnt = nc;
        }
        uint32_t B = (uint32_t)((g << 5) + 31);
        for (int b = (g << 5); b < (g << 5) + 32; ++b) {
            uint32_t nc = cnt + hc[b];
            if (nc > jrank) { B = (uint32_t)b; break; }
            cnt = nc;
        }
        scr_u[2] = B; scr_u[3] = cnt;
    }
    __syncthreads();
}

__global__ __launch_bounds__(1024, 1)
void eps_threshold_kernel(const float* __restrict__ logprob,
                          float* __restrict__ out, int V) {
    extern __shared__ unsigned char smem[];
    float*    hsum  = (float*)(smem + OFF_HSUM);
    uint32_t* hcnt  = (uint32_t*)(smem + OFF_HCNT);
    uint32_t* hcnt0 = (uint32_t*)(smem + OFF_HCNT0);
    double*   gsum  = (double*)(smem + OFF_GSUM);
    uint32_t* gcnt  = (uint32_t*)(smem + OFF_GCNT);
    uint32_t* scr_u = (uint32_t*)(smem + OFF_SCRU);
    double*   scr_d = (double*)(smem + OFF_SCRD);
    uint32_t* vals  = (uint32_t*)(smem + OFF_VALS);
    uint4*    vals4 = (uint4*)(smem + OFF_VALS);

    const int tid = threadIdx.x;
    const int nt  = blockDim.x;
    const int row = blockIdx.x;
    const int V4  = V >> 2;

    // ---- Stage 0: async global -> LDS DMA of the whole row -------------------
    const uint64_t base = (uint64_t)(const void*)(logprob + (size_t)row * (size_t)V);
    for (int i = tid; i < V; i += nt) {
        uint32_t la = (uint32_t)(uintptr_t)(&vals[i]);
        uint64_t ga = base + (uint64_t)i * 4ull;
        asm volatile("global_load_async_to_lds_b32 %0, %1, off"
                     :: "v"(la), "v"(ga) : "memory");
    }
    // Zero round-0 histograms while the DMA is in flight (free overlap).
    for (int b = tid; b < 4096; b += nt) { hcnt[b] = 0u; hsum[b] = 0.0f; }
    asm volatile("s_wait_asynccnt 0" ::: "memory");
    __syncthreads();

    // ---- Fused sweep: exp() in place + 12-bit count/sum histogram ------------
    for (int c = tid; c < V4; c += nt) {
        uint4 kv = vals4[c];
        float p[4];
        p[0] = expf(__uint_as_float(kv.x));
        p[1] = expf(__uint_as_float(kv.y));
        p[2] = expf(__uint_as_float(kv.z));
        p[3] = expf(__uint_as_float(kv.w));
        kv.x = __float_as_uint(p[0]); kv.y = __float_as_uint(p[1]);
        kv.z = __float_as_uint(p[2]); kv.w = __float_as_uint(p[3]);
        vals4[c] = kv;
        uint32_t ks[4] = {kv.x, kv.y, kv.z, kv.w};
#pragma unroll
        for (int e = 0; e < 4; ++e) {
            uint32_t b = ks[e] >> 20;
            atomicAdd(&hcnt[b], 1u);
            atomicAdd(&hsum[b], p[e]);
        }
    }
    for (int i = (V4 << 2) + tid; i < V; i += nt) {
        float pv = expf(__uint_as_float(vals[i]));
        uint32_t k = __float_as_uint(pv);
        vals[i] = k;
        atomicAdd(&hcnt[k >> 20], 1u);
        atomicAdd(&hsum[k >> 20], pv);
    }
    __syncthreads();

    // Preserve round-0 counts: stage-2 round 0 is exactly this histogram.
    for (int b = tid; b < 4096; b += nt) hcnt0[b] = hcnt[b];

    // ---- Stage 1: weighted radix-select (cumsum > EPS crossing) --------------
    scan_cross_sum(hsum, hcnt, gsum, gcnt, scr_u, scr_d, 4096, 0.0, tid);
    uint32_t prefix  = scr_u[0] << 20;
    double   baseSum = scr_d[0];
    uint32_t m       = scr_u[1];

    for (int r = 1; r < 3; ++r) {
        const int      shift  = (r == 1) ? 10 : 0;
        const uint32_t maskHi = (r == 1) ? 0xFFF00000u : 0xFFFFFC00u;
        for (int b = tid; b < 1024; b += nt) { hcnt[b] = 0u; hsum[b] = 0.0f; }
        __syncthreads();
        for (int c = tid; c < V4; c += nt) {
            uint4 kv = vals4[c];
            uint32_t ks[4] = {kv.x, kv.y, kv.z, kv.w};
#pragma unroll
            for (int e = 0; e < 4; ++e) {
                uint32_t k = ks[e];
                if ((k & maskHi) == prefix) {
                    uint32_t b = (k >> shift) & 1023u;
                    atomicAdd(&hcnt[b], 1u);
                    atomicAdd(&hsum[b], __uint_as_float(k));
                }
            }
        }
        for (int i = (V4 << 2) + tid; i < V; i += nt) {
            uint32_t k = vals[i];
            if ((k & maskHi) == prefix) {
                atomicAdd(&hcnt[(k >> shift) & 1023u], 1u);
                atomicAdd(&hsum[(k >> shift) & 1023u], __uint_as_float(k));
            }
        }
        __syncthreads();
        scan_cross_sum(hsum, hcnt, gsum, gcnt, scr_u, scr_d, 1024, baseSum, tid);
        prefix |= scr_u[0] << shift;
        baseSum = scr_d[0];
        m       = scr_u[1];
    }

    const uint32_t vbits = prefix;                 // exact bits of crossing element
    const double   vd    = (double)__uint_as_float(vbits);

    // Which duplicate copy crosses EPS (stable argsort: ties by ascending index)
    int cdup = 0;
    while ((cdup + 1) < (int)m && (baseSum + (double)(cdup + 1) * vd) <= EPS_D) ++cdup;

    // jrank = cdup-th smallest original index among elements equal to vbits
    int last = -1;
    for (int it = 0; it <= cdup; ++it) {
        if (tid == 0) scr_u[4] = 0x7FFFFFFFu;
        __syncthreads();
        for (int c = tid; c < V4; c += nt) {
            uint4 kv = vals4[c];
            const int ib = c << 2;
            if (kv.x == vbits && ib + 0 > last) atomicMin((int*)&scr_u[4], ib + 0);
            if (kv.y == vbits && ib + 1 > last) atomicMin((int*)&scr_u[4], ib + 1);
            if (kv.z == vbits && ib + 2 > last) atomicMin((int*)&scr_u[4], ib + 2);
            if (kv.w == vbits && ib + 3 > last) atomicMin((int*)&scr_u[4], ib + 3);
        }
        for (int i = (V4 << 2) + tid; i < V; i += nt)
            if (vals[i] == vbits && i > last) atomicMin((int*)&scr_u[4], i);
        __syncthreads();
        last = (int)scr_u[4];
        __syncthreads();
    }
    const uint32_t jrank = (uint32_t)last;

    // ---- Stage 2: count radix-select at rank jrank ---------------------------
    // Round 0 reuses the saved unprefixed 12-bit count histogram (no sweep).
    scan_rank(hcnt0, gcnt, scr_u, 4096, 0u, jrank, tid);
    uint32_t prefix2 = scr_u[2] << 20;
    uint32_t baseCnt = scr_u[3];

    for (int r = 1; r < 3; ++r) {
        const int      shift  = (r == 1) ? 10 : 0;
        const uint32_t maskHi = (r == 1) ? 0xFFF00000u : 0xFFFFFC00u;
        for (int b = tid; b < 1024; b += nt) hcnt[b] = 0u;
        __syncthreads();
        for (int c = tid; c < V4; c += nt) {
            uint4 kv = vals4[c];
            uint32_t ks[4] = {kv.x, kv.y, kv.z, kv.w};
#pragma unroll
            for (int e = 0; e < 4; ++e) {
                uint32_t k = ks[e];
                if ((k & maskHi) == prefix2)
                    atomicAdd(&hcnt[(k >> shift) & 1023u], 1u);
            }
        }
        for (int i = (V4 << 2) + tid; i < V; i += nt) {
            uint32_t k = vals[i];
            if ((k & maskHi) == prefix2)
                atomicAdd(&hcnt[(k >> shift) & 1023u], 1u);
        }
        __syncthreads();
        scan_rank(hcnt, gcnt, scr_u, 1024, baseCnt, jrank, tid);
        prefix2 |= scr_u[2] << shift;
        baseCnt  = scr_u[3];
    }

    if (tid == 0) {
        out[row] = logf(__uint_as_float(prefix2));
    }
}

extern "C" void kernel_launch(void* const* d_in, const int* in_sizes, int n_in,
                              void* d_out, int out_size, void* d_ws, size_t ws_size,
                              hipStream_t stream) {
    (void)n_in; (void)d_ws; (void)ws_size;
    const float* logprob = (const float*)d_in[0];
    float* out = (float*)d_out;

    const int B = out_size;                        // 2048 rows
    const int V = (B > 0) ? (in_sizes[0] / B) : 0; // 50257 vocab

    // dynamic LDS: 50,816 B of histograms/scratch + V*4 B row ~= 252 KB (< 320 KB/WGP)
    const size_t shmem = (size_t)OFF_VALS + (size_t)V * 4u + 16u;

    eps_threshold_kernel<<<B, 1024, shmem, stream>>>(logprob, out, V);
}